// ExtractModel_60189671686714
// MI455X (gfx1250) — compile-verified
//
#include <hip/hip_runtime.h>
#include <hip/hip_bf16.h>
#include <math.h>

#define B_      16
#define L_      50
#define LU_     60
#define KU_     100
#define DIM_    60
#define V_      2000
#define MAXVL_  10
#define MIN_WL_ 4
#define MAX_WL_ 10
#define NE_     7
#define IDC     3.5f
#define NEGV    -9999.9f
#define CW      0.1f
#define LOGU    -4.605170185988091f   // log(0.01)

typedef float v2f __attribute__((ext_vector_type(2)));
typedef float v8f __attribute__((ext_vector_type(8)));

// ---------------------------------------------------------------------------
// K1: ku_char_repr = alw(60x100) @ U(100x60)        [WMMA f32 16x16x4]
//     logits      = ku(60x60)   @ U^T(60x100)       [WMMA f32 16x16x4]
//     char_log_probs = log_softmax(logits) -> ws ;  alignment = exp(lp) -> out
// Single block, 256 threads = 8 waves.
// ---------------------------------------------------------------------------
__global__ __launch_bounds__(256) void k1_char_repr(
    const float* __restrict__ U,        // KU x DIM
    const float* __restrict__ alw,      // LU x KU
    float* __restrict__ ku_ws,          // LU x DIM
    float* __restrict__ clp_ws,         // LU x KU
    float* __restrict__ align_out)      // LU x KU
{
    __shared__ float ku_s[LU_ * DIM_];
    __shared__ float lg_s[LU_ * KU_];
    const int tid  = threadIdx.x;
    const int wave = tid >> 5;
    const int lane = tid & 31;
    const int half = lane >> 4;     // which K-half this lane holds
    const int lm   = lane & 15;

    // ---- GEMM1: 4x4 tiles of 16x16, K=100 (25 wmma steps) ----
    for (int t = wave; t < 16; t += 8) {
        const int mi = t >> 2, nj = t & 3;
        const int m  = mi * 16 + lm;
        const int n  = nj * 16 + lm;
        const int mc = (m < LU_)  ? m : (LU_ - 1);   // clamped (loads stay uniform)
        const int nc = (n < DIM_) ? n : (DIM_ - 1);
        v8f acc = {};
        for (int ks = 0; ks < 25; ++ks) {
            const int k0 = ks * 4 + half * 2;        // k0+1 <= 99
            v2f a, b;
            const float a0 = alw[mc * KU_ + k0];
            const float a1 = alw[mc * KU_ + k0 + 1];
            a.x = (m < LU_) ? a0 : 0.0f;
            a.y = (m < LU_) ? a1 : 0.0f;
            const float b0 = U[k0 * DIM_ + nc];
            const float b1 = U[(k0 + 1) * DIM_ + nc];
            b.x = (n < DIM_) ? b0 : 0.0f;
            b.y = (n < DIM_) ? b1 : 0.0f;
            acc = __builtin_amdgcn_wmma_f32_16x16x4_f32(
                false, a, false, b, (short)0, acc, false, false);
        }
        for (int r = 0; r < 8; ++r) {
            const int mm = mi * 16 + r + half * 8;
            const int nn = nj * 16 + lm;
            if (mm < LU_ && nn < DIM_) {
                ku_s[mm * DIM_ + nn]  = acc[r];
                ku_ws[mm * DIM_ + nn] = acc[r];
            }
        }
    }
    __syncthreads();

    // ---- GEMM2: 4x7 tiles, K=60 (15 wmma steps); B[d,n] = U[n*DIM + d] ----
    for (int t = wave; t < 28; t += 8) {
        const int mi = t / 7, nj = t % 7;
        const int m  = mi * 16 + lm;
        const int n  = nj * 16 + lm;
        const int mc = (m < LU_) ? m : (LU_ - 1);
        const int nc = (n < KU_) ? n : (KU_ - 1);
        v8f acc = {};
        for (int ks = 0; ks < 15; ++ks) {
            const int k0 = ks * 4 + half * 2;        // k0+1 <= 59
            v2f a, b;
            const float a0 = ku_s[mc * DIM_ + k0];
            const float a1 = ku_s[mc * DIM_ + k0 + 1];
            a.x = (m < LU_) ? a0 : 0.0f;
            a.y = (m < LU_) ? a1 : 0.0f;
            const float b0 = U[nc * DIM_ + k0];
            const float b1 = U[nc * DIM_ + k0 + 1];
            b.x = (n < KU_) ? b0 : 0.0f;
            b.y = (n < KU_) ? b1 : 0.0f;
            acc = __builtin_amdgcn_wmma_f32_16x16x4_f32(
                false, a, false, b, (short)0, acc, false, false);
        }
        for (int r = 0; r < 8; ++r) {
            const int mm = mi * 16 + r + half * 8;
            const int nn = nj * 16 + lm;
            if (mm < LU_ && nn < KU_) lg_s[mm * KU_ + nn] = acc[r];
        }
    }
    __syncthreads();

    // ---- row-wise log_softmax over KU ----
    if (tid < LU_) {
        const float* row = &lg_s[tid * KU_];
        float mx = row[0];
        for (int k = 1; k < KU_; ++k) mx = fmaxf(mx, row[k]);
        float s = 0.0f;
        for (int k = 0; k < KU_; ++k) s += expf(row[k] - mx);
        const float lse = mx + logf(s);
        for (int k = 0; k < KU_; ++k) {
            const float lp = row[k] - lse;
            clp_ws[tid * KU_ + k]    = lp;
            align_out[tid * KU_ + k] = expf(lp);
        }
    }
}

// ---------------------------------------------------------------------------
// K2: one block per (b,t): conv(width 3) -> word_repr row, ctx logits (100),
//     log_softmax, pos_lp[b,t,k] = char_lp[uid][k] + 0.1*ctx_lp[k]
// ---------------------------------------------------------------------------
__global__ __launch_bounds__(128) void k2_poslp(
    const int*   __restrict__ uid,      // B x L
    const float* __restrict__ U,        // KU x DIM
    const float* __restrict__ conv_w,   // DIM x DIM x 3 (O,I,H)
    const float* __restrict__ conv_b,   // DIM
    const float* __restrict__ ku_ws,    // LU x DIM
    const float* __restrict__ clp_ws,   // LU x KU
    float* __restrict__ pos_lp)         // B x L x KU
{
    const int b   = blockIdx.x / L_;
    const int t   = blockIdx.x % L_;
    const int tid = threadIdx.x;
    __shared__ float e3[3 * DIM_];
    __shared__ float wr[DIM_];
    __shared__ float lg[KU_];
    __shared__ float lse_s;

    for (int i = tid; i < 3 * DIM_; i += 128) {
        const int r = i / DIM_, d = i % DIM_;
        const int t2 = t + r - 1;
        float v = 0.0f;
        if (t2 >= 0 && t2 < L_) v = ku_ws[uid[b * L_ + t2] * DIM_ + d];
        e3[i] = v;
    }
    __syncthreads();

    if (tid < DIM_) {                               // o = tid
        float acc = conv_b[tid];
        for (int r = 0; r < 3; ++r)
            for (int d = 0; d < DIM_; ++d)
                acc = fmaf(conv_w[tid * DIM_ * 3 + d * 3 + r], e3[r * DIM_ + d], acc);
        wr[tid] = acc;
    }
    __syncthreads();

    if (tid < KU_) {                                // k = tid
        float acc = 0.0f;
        for (int d = 0; d < DIM_; ++d)
            acc = fmaf(wr[d], U[tid * DIM_ + d], acc);
        lg[tid] = acc;
    }
    __syncthreads();

    if (tid == 0) {
        float mx = lg[0];
        for (int k = 1; k < KU_; ++k) mx = fmaxf(mx, lg[k]);
        float s = 0.0f;
        for (int k = 0; k < KU_; ++k) s += expf(lg[k] - mx);
        lse_s = mx + logf(s);
    }
    __syncthreads();

    if (tid < KU_) {
        const int u = uid[b * L_ + t];
        pos_lp[(b * L_ + t) * KU_ + tid] =
            clp_ws[u * KU_ + tid] + CW * (lg[tid] - lse_s);
    }
}

// ---------------------------------------------------------------------------
// K3: max-plus DP. One block per (b,l); 10 pos_lp rows in LDS; DP front and
//     vocab ids in registers. Packed-u64 atomicMax argmax per batch.
// ---------------------------------------------------------------------------
__device__ __forceinline__ unsigned int fkey(float f) {
    const unsigned int b = __float_as_uint(f);
    return (b & 0x80000000u) ? ~b : (b | 0x80000000u);
}

__global__ void k_init(unsigned long long* best) {
    if (threadIdx.x < B_) best[threadIdx.x] = 0ull;
}

__global__ __launch_bounds__(256) void k3_dp(
    const float* __restrict__ pos_lp,       // B x L x KU
    const int*   __restrict__ lengths,      // B
    const int*   __restrict__ vocab_ids,    // V x MAXVL
    const int*   __restrict__ vocab_lengths,// V
    unsigned long long* __restrict__ best)  // B
{
    const int b   = blockIdx.x / L_;
    const int l   = blockIdx.x % L_;
    const int tid = threadIdx.x;
    __shared__ float rows[MAX_WL_ * KU_];
    __shared__ float rv[256];
    __shared__ int   ri[256];

    for (int i = tid; i < MAX_WL_ * KU_; i += 256) {
        const int r = i / KU_, k = i % KU_;
        const int t = l + r;
        rows[i] = (t < L_) ? pos_lp[(b * L_ + t) * KU_ + k] : NEGV;
    }
    const int len = lengths[b];
    __syncthreads();

    float bv = -3.4e38f;
    int   bi = 0x7FFFFFFF;

    for (int v = tid; v < V_; v += 256) {
        int kid[MAXVL_];
#pragma unroll
        for (int j = 0; j < MAXVL_; ++j) kid[j] = vocab_ids[v * MAXVL_ + j];
        const int vl = vocab_lengths[v];

        float prev[MAXVL_ + 1];
#pragma unroll
        for (int j = 0; j <= MAXVL_; ++j) prev[j] = -IDC * (float)j;

#pragma unroll
        for (int i = 1; i <= MAX_WL_; ++i) {
            const float* srow = &rows[(i - 1) * KU_];
            float carry = prev[0];
            float leftv = -IDC * (float)i;
            prev[0] = leftv;
#pragma unroll
            for (int j = 1; j <= MAXVL_; ++j) {
                const float pj   = prev[j];
                const float diag = carry + srow[kid[j - 1]];
                const float m = fmaxf(diag, fmaxf(pj - IDC, leftv - IDC));
                prev[j] = m; leftv = m; carry = pj;
            }
            if (i >= MIN_WL_) {
                // prev[vl] without dynamic register indexing
                float llv = prev[MIN_WL_];
#pragma unroll
                for (int j = MIN_WL_ + 1; j <= MAXVL_; ++j)
                    if (vl == j) llv = prev[j];
                float sc = llv + (float)(len - i) * LOGU;
                if (l + i > len) sc += NEGV;               // non-viable
                const int e   = i - MIN_WL_;
                const int idx = (l * NE_ + e) * V_ + v;
                if (sc > bv || (sc == bv && idx < bi)) { bv = sc; bi = idx; }
            }
        }
    }

    rv[tid] = bv; ri[tid] = bi;
    __syncthreads();
    for (int s = 128; s > 0; s >>= 1) {
        if (tid < s) {
            const float ov = rv[tid + s]; const int oi = ri[tid + s];
            if (ov > rv[tid] || (ov == rv[tid] && oi < ri[tid])) { rv[tid] = ov; ri[tid] = oi; }
        }
        __syncthreads();
    }
    if (tid == 0) {
        const unsigned long long p =
            ((unsigned long long)fkey(rv[0]) << 32) |
            (unsigned long long)(0xFFFFFFFFu - (unsigned int)ri[0]);
        atomicMax(&best[b], p);
    }
}

__global__ void k4_final(const unsigned long long* __restrict__ best,
                         float* __restrict__ out)
{
    const int b = threadIdx.x;
    if (b < B_) {
        const unsigned long long p = best[b];
        const unsigned int key  = (unsigned int)(p >> 32);
        const unsigned int bits = (key & 0x80000000u) ? (key ^ 0x80000000u) : ~key;
        const float val = __uint_as_float(bits);
        const unsigned int idx = 0xFFFFFFFFu - (unsigned int)(p & 0xFFFFFFFFu);
        const int vv    = (int)(idx % V_);
        const int rest  = (int)(idx / V_);
        const int e     = rest % NE_;
        const int start = rest / NE_;
        const int end   = start + MIN_WL_ + e - 1;
        out[b]          = (float)start;
        out[B_ + b]     = (float)end;
        out[2 * B_ + b] = val;
        out[3 * B_ + b] = (float)vv;
    }
}

// ---------------------------------------------------------------------------
extern "C" void kernel_launch(void* const* d_in, const int* in_sizes, int n_in,
                              void* d_out, int out_size, void* d_ws, size_t ws_size,
                              hipStream_t stream) {
    (void)in_sizes; (void)n_in; (void)out_size; (void)ws_size;
    const int*   uid           = (const int*)d_in[0];
    const int*   lengths       = (const int*)d_in[1];
    const float* unit_repr     = (const float*)d_in[2];
    const float* alw           = (const float*)d_in[3];
    const float* conv_w        = (const float*)d_in[4];
    const float* conv_b        = (const float*)d_in[5];
    const int*   vocab_ids     = (const int*)d_in[6];
    const int*   vocab_lengths = (const int*)d_in[7];
    float* out = (float*)d_out;

    float* ws     = (float*)d_ws;
    float* ku_ws  = ws;                 // 3600 floats
    float* clp_ws = ws + 3600;          // 6000 floats
    float* pos_lp = ws + 9600;          // 80000 floats
    unsigned long long* best = (unsigned long long*)(ws + 89600);  // 16 u64, 8B aligned

    k1_char_repr<<<1, 256, 0, stream>>>(unit_repr, alw, ku_ws, clp_ws, out + 4 * B_);
    k2_poslp<<<B_ * L_, 128, 0, stream>>>(uid, unit_repr, conv_w, conv_b, ku_ws, clp_ws, pos_lp);
    k_init<<<1, 32, 0, stream>>>(best);
    k3_dp<<<B_ * L_, 256, 0, stream>>>(pos_lp, lengths, vocab_ids, vocab_lengths, best);
    k4_final<<<1, 32, 0, stream>>>(best, out);
}